// QueryMemoryBank_62397284876817
// MI455X (gfx1250) — compile-verified
//
#include <hip/hip_runtime.h>
#include <math.h>

#define EMBED_DIM   512
#define MEMORY_SIZE 262144
#define NEG_INF     (-1e30f)

typedef __attribute__((ext_vector_type(2))) float v2f;
typedef __attribute__((ext_vector_type(8))) float v8f;

constexpr int ROWS_PER_BLOCK = 128;                          // 8 waves x 16 rows
constexpr int NBLOCKS_A      = MEMORY_SIZE / ROWS_PER_BLOCK; // 2048
constexpr int NBLOCKS_C      = MEMORY_SIZE / 256;            // 1024

// workspace layout (float offsets)
constexpr size_t WS_LOGITS   = 0;                        // [M] logits -> weights (in place)
constexpr size_t WS_BLOCKMAX = MEMORY_SIZE;              // [NBLOCKS_A]
constexpr size_t WS_BLOCKSUM = WS_BLOCKMAX + NBLOCKS_A;  // [NBLOCKS_C]
constexpr size_t WS_SCALARS  = WS_BLOCKSUM + NBLOCKS_C;  // [gmax, any_valid, invS]

// ---------------------------------------------------------------------------
// Pass 1: sims via V_WMMA_F32_16X16X4_F32. Each wave owns 16 memory rows.
//
// Per 8-dim step, one global_load_b128 per lane feeds TWO WMMAs. K-slot->dim
// mapping (consistent between A and B):
//   wmma1: K0=dim k,   K1=k+1, K2=k+4, K3=k+5   (from float4 .x/.y)
//   wmma2: K0=dim k+2, K1=k+3, K2=k+6, K3=k+7   (from float4 .z/.w)
// A layout (ISA): vgpr0/half0=K0, vgpr1/half0=K1, vgpr0/half1=K2, vgpr1/half1=K3.
// B carries the query only in column N==0: lanes 0/16 walk qs[] with stride 8;
// all other lanes do a broadcast ds_load of a zeroed pad (stride 0) -- fully
// unconditional, no exec masking in the loop.
// D (16x16): column N=0 -> lane 0 holds M=0..7 in acc[0..7], lane 16 M=8..15.
// ---------------------------------------------------------------------------
__global__ void __launch_bounds__(256)
sims_wmma_kernel(const float* __restrict__ q,
                 const float* __restrict__ qcard,
                 const float* __restrict__ E,
                 const float* __restrict__ mcard,
                 const unsigned char* __restrict__ used,
                 float* __restrict__ logits,
                 float* __restrict__ blockmax)
{
    __shared__ float qs[EMBED_DIM + 4];   // query + 4-float zero pad
    __shared__ float wmax[16];

    const int tid = threadIdx.x;
    for (int i = tid; i < EMBED_DIM; i += 256) qs[i] = q[i];
    if (tid < 4) qs[EMBED_DIM + tid] = 0.0f;
    __syncthreads();

    const int wave = tid >> 5;
    const int lane = tid & 31;
    const int half = lane >> 4;     // 0: lanes 0-15, 1: lanes 16-31
    const int l15  = lane & 15;
    const int rowBase = blockIdx.x * ROWS_PER_BLOCK + wave * 16;

    const float* arow = E + (size_t)(rowBase + l15) * EMBED_DIM + 4 * half;
    const bool bactive = (l15 == 0);             // only column N==0 carries q
    int       qoff = bactive ? (4 * half) : EMBED_DIM;  // inactive -> zero pad
    const int qinc = bactive ? 8 : 0;

    v8f acc = {};
    #pragma unroll 4
    for (int k = 0; k < EMBED_DIM; k += 8) {
        float4 av = *(const float4*)(arow + k);   // global_load_b128, 16B aligned
        float4 bv = *(const float4*)&qs[qoff];    // ds_load_b128 (broadcast for N!=0)
        qoff += qinc;

        v2f a0; a0.x = av.x; a0.y = av.y;
        v2f b0; b0.x = bv.x; b0.y = bv.y;
        acc = __builtin_amdgcn_wmma_f32_16x16x4_f32(
                  false, a0, false, b0, (short)0, acc, false, false);

        v2f a1; a1.x = av.z; a1.y = av.w;
        v2f b1; b1.x = bv.z; b1.y = bv.w;
        acc = __builtin_amdgcn_wmma_f32_16x16x4_f32(
                  false, a1, false, b1, (short)0, acc, false, false);
    }

    const float qc = qcard[0];          // mean over 1 element
    float lmax = NEG_INF;
    if (bactive) {
        const int mbase = rowBase + half * 8;
        #pragma unroll
        for (int i = 0; i < 8; ++i) {
            const int m = mbase + i;
            const float sim = acc[i];
            const bool valid = (used[m] != 0) && (sim >= 0.9f) &&
                               (fabsf(qc - mcard[m]) <= 1.0f);
            const float logit = valid ? sim * 10.0f : NEG_INF;   // sims / 0.1
            logits[m] = logit;
            lmax = fmaxf(lmax, logit);
        }
        wmax[wave * 2 + half] = lmax;
    }
    __syncthreads();
    if (tid == 0) {
        float m = NEG_INF;
        #pragma unroll
        for (int i = 0; i < 16; ++i) m = fmaxf(m, wmax[i]);
        blockmax[blockIdx.x] = m;
    }
}

// ---------------------------------------------------------------------------
// Pass 2: reduce 2048 partial maxima -> gmax, any_valid
// ---------------------------------------------------------------------------
__global__ void __launch_bounds__(256)
reduce_max_kernel(const float* __restrict__ blockmax, float* __restrict__ scalars)
{
    __shared__ float s[256];
    float m = NEG_INF;
    for (int i = threadIdx.x; i < NBLOCKS_A; i += 256) m = fmaxf(m, blockmax[i]);
    s[threadIdx.x] = m;
    __syncthreads();
    for (int st = 128; st > 0; st >>= 1) {
        if (threadIdx.x < st) s[threadIdx.x] = fmaxf(s[threadIdx.x], s[threadIdx.x + st]);
        __syncthreads();
    }
    if (threadIdx.x == 0) {
        scalars[0] = s[0];
        scalars[1] = (s[0] > -1e29f) ? 1.0f : 0.0f;   // any_valid
    }
}

// ---------------------------------------------------------------------------
// Pass 3: weights = exp(logit - gmax) (masked entries underflow to exactly 0)
// ---------------------------------------------------------------------------
__global__ void __launch_bounds__(256)
weights_kernel(float* __restrict__ logits,            // in-place -> weights
               const float* __restrict__ scalars,
               float* __restrict__ blocksum)
{
    __shared__ float s[256];
    const int i = blockIdx.x * 256 + threadIdx.x;
    const float gmax = scalars[0];
    const float anyv = scalars[1];
    const float l = logits[i];
    const float w = (anyv != 0.0f) ? __expf(l - gmax) : 0.0f;
    logits[i] = w;
    s[threadIdx.x] = w;
    __syncthreads();
    for (int st = 128; st > 0; st >>= 1) {
        if (threadIdx.x < st) s[threadIdx.x] += s[threadIdx.x + st];
        __syncthreads();
    }
    if (threadIdx.x == 0) blocksum[blockIdx.x] = s[0];
}

// ---------------------------------------------------------------------------
// Pass 4: total sum -> invS; also zero-initialize the output vector
// ---------------------------------------------------------------------------
__global__ void __launch_bounds__(256)
reduce_sum_kernel(const float* __restrict__ blocksum,
                  float* __restrict__ scalars,
                  float* __restrict__ out)
{
    __shared__ float s[256];
    float t = 0.0f;
    for (int i = threadIdx.x; i < NBLOCKS_C; i += 256) t += blocksum[i];
    s[threadIdx.x] = t;
    __syncthreads();
    for (int st = 128; st > 0; st >>= 1) {
        if (threadIdx.x < st) s[threadIdx.x] += s[threadIdx.x + st];
        __syncthreads();
    }
    if (threadIdx.x == 0)
        scalars[2] = (s[0] > 0.0f) ? 1.0f / s[0] : 0.0f;   // invS (0 if no valid)
    out[threadIdx.x]       = 0.0f;
    out[threadIdx.x + 256] = 0.0f;
}

// ---------------------------------------------------------------------------
// Pass 5: out[d] = sum_m (w_m * invS) * E[m][d]; coalesced float2 per lane,
// rows with zero weight (the overwhelming majority after softmax) are skipped.
// ---------------------------------------------------------------------------
__global__ void __launch_bounds__(256)
weighted_sum_kernel(const float* __restrict__ E,
                    const float* __restrict__ weights,
                    const float* __restrict__ scalars,
                    float* __restrict__ out)
{
    __shared__ float wl[ROWS_PER_BLOCK];
    const int tid = threadIdx.x;
    const int rowBase = blockIdx.x * ROWS_PER_BLOCK;
    if (tid < ROWS_PER_BLOCK) wl[tid] = weights[rowBase + tid];
    __syncthreads();

    const float invS = scalars[2];
    float2 acc = make_float2(0.0f, 0.0f);
    const float2* E2 = (const float2*)E;    // thread owns dims 2*tid, 2*tid+1

    for (int r = 0; r < ROWS_PER_BLOCK; ++r) {
        const float w = wl[r] * invS;
        if (w != 0.0f) {
            float2 e = E2[(size_t)(rowBase + r) * (EMBED_DIM / 2) + tid];
            acc.x += w * e.x;
            acc.y += w * e.y;
        }
    }
    atomicAdd(&out[2 * tid],     acc.x);
    atomicAdd(&out[2 * tid + 1], acc.y);
}

// ---------------------------------------------------------------------------
extern "C" void kernel_launch(void* const* d_in, const int* in_sizes, int n_in,
                              void* d_out, int out_size, void* d_ws, size_t ws_size,
                              hipStream_t stream)
{
    const float*         q     = (const float*)d_in[0];         // [1,512]
    const float*         qcard = (const float*)d_in[1];         // [1]
    const float*         E     = (const float*)d_in[2];         // [M,512]
    const float*         mcard = (const float*)d_in[3];         // [M]
    const unsigned char* used  = (const unsigned char*)d_in[4]; // [M] bool

    float* out = (float*)d_out;
    float* ws  = (float*)d_ws;
    float* logits   = ws + WS_LOGITS;
    float* blockmax = ws + WS_BLOCKMAX;
    float* blocksum = ws + WS_BLOCKSUM;
    float* scalars  = ws + WS_SCALARS;

    sims_wmma_kernel   <<<NBLOCKS_A, 256, 0, stream>>>(q, qcard, E, mcard, used,
                                                       logits, blockmax);
    reduce_max_kernel  <<<1,         256, 0, stream>>>(blockmax, scalars);
    weights_kernel     <<<NBLOCKS_C, 256, 0, stream>>>(logits, scalars, blocksum);
    reduce_sum_kernel  <<<1,         256, 0, stream>>>(blocksum, scalars, out);
    weighted_sum_kernel<<<NBLOCKS_A, 256, 0, stream>>>(E, logits, scalars, out);
}